// HGConstruct_57415122813702
// MI455X (gfx1250) — compile-verified
//
#include <hip/hip_runtime.h>
#include <stdint.h>

#define NPTS        16384
#define BS          1024
#define CHUNK       16            // NPTS / BS
#define EPSV        0.1f
#define MIN_SAMPLES 5
#define NC_MAX      512
#define BATCH       8
#define DIMS        64

typedef unsigned int u32x4 __attribute__((ext_vector_type(4)));
typedef int          i32x8 __attribute__((ext_vector_type(8)));
typedef int          i32x4 __attribute__((ext_vector_type(4)));
typedef float        f32x4 __attribute__((ext_vector_type(4)));

__device__ __forceinline__ int lower_bound_lds(const float* a, float v) {
    int lo = 0, hi = NPTS;
    while (lo < hi) { int m = (lo + hi) >> 1; if (a[m] < v) lo = m + 1; else hi = m; }
    return lo;
}
__device__ __forceinline__ int upper_bound_lds(const float* a, float v) {
    int lo = 0, hi = NPTS;
    while (lo < hi) { int m = (lo + hi) >> 1; if (a[m] <= v) lo = m + 1; else hi = m; }
    return lo;
}

// One block per (batch, dim) pair: full 1-D DBSCAN in LDS.
__global__ __launch_bounds__(BS) void dbscan1d_kernel(const float* __restrict__ feat,
                                                      int* __restrict__ labels,
                                                      int* __restrict__ ncl) {
    extern __shared__ char smem[];
    float* skey  = (float*)smem;                        // 64 KB sorted keys
    int*   slab  = (int*)(smem + (size_t)NPTS * 4);     // 64 KB labels (sorted order)
    float* sbufF = (float*)(smem + (size_t)NPTS * 8);   // 4 KB scan buffer
    int*   sbufI = (int*)sbufF;

    const int tid = threadIdx.x;
    const int bd  = blockIdx.x;
    const int b   = bd >> 6, d = bd & 63;
    const float INF = __builtin_inff();

    // ---- TDM: DMA the strided column feat[b, :, d] into LDS ----
    // Element (n,d) lives at base + (n*64 + d)*4 : tile of 16384 rows x 1 elem,
    // row stride 64 elements, packed contiguously into skey[].
    {
        uint64_t ga      = (uint64_t)(size_t)feat + ((uint64_t)b * NPTS * DIMS + (uint64_t)d) * 4ull;
        unsigned lds_off = (unsigned)(size_t)skey;      // low 32 bits of LDS aperture addr = offset
        if (tid == 0) {
            u32x4 g0 = { 1u,                                        // count=1 (valid), no gather
                         lds_off,                                   // D#.lds_addr
                         (unsigned)(ga & 0xFFFFFFFFu),              // global_addr[31:0]
                         (unsigned)(((ga >> 32) & 0x1FFFFFFu) | (2u << 30)) }; // addr[56:32] | type=2
            i32x8 g1 = { (int)0x00020000u,   // workgroup_mask=0, data_size=2 (4B), no flags
                         (int)0x00010000u,   // tensor_dim0 = 1   (bits 79:48 low half)
                         (int)0x40000000u,   // tensor_dim0 hi=0, tensor_dim1 lo16 = 16384
                         (int)0x00010000u,   // tensor_dim1 hi=0, tile_dim0 = 1
                         (int)0x00004000u,   // tile_dim1 = 16384, tile_dim2 = 0
                         64,                 // tensor_dim0_stride = 64 elements (row->row)
                         0, 0 };             // stride hi / tensor_dim1_stride = 0 (2-D)
            i32x4 gz4 = { 0, 0, 0, 0 };
            i32x8 gz8 = { 0, 0, 0, 0, 0, 0, 0, 0 };
#if __has_builtin(__builtin_amdgcn_tensor_load_to_lds)
            __builtin_amdgcn_tensor_load_to_lds(g0, g1, gz4, gz4, gz8, 0);
#endif
        }
#if __has_builtin(__builtin_amdgcn_s_wait_tensorcnt)
        __builtin_amdgcn_s_wait_tensorcnt(0);
#endif
        __syncthreads();
#if !__has_builtin(__builtin_amdgcn_tensor_load_to_lds)
        for (int n = tid; n < NPTS; n += BS)
            skey[n] = feat[((size_t)b * NPTS + n) * DIMS + d];
        __syncthreads();
#endif
    }

    // ---- bitonic sort of 16384 keys in LDS ----
    for (unsigned k = 2; k <= NPTS; k <<= 1) {
        for (unsigned j = k >> 1; j > 0; j >>= 1) {
            for (unsigned i = tid; i < NPTS; i += BS) {
                unsigned p = i ^ j;
                if (p > i) {
                    float a = skey[i], c = skey[p];
                    bool asc = ((i & k) == 0);
                    if (asc ? (a > c) : (a < c)) { skey[i] = c; skey[p] = a; }
                }
            }
            __syncthreads();
        }
    }

    // ---- core test (two binary searches per point), chunk of 16 per thread ----
    const int g0i = tid * CHUNK;
    unsigned mask = 0;
    float cmax = -INF, cminR = INF;       // chunk reductions over core values
#pragma unroll
    for (int i = 0; i < CHUNK; ++i) {
        float xs = skey[g0i + i];
        int hi = upper_bound_lds(skey, xs + EPSV);
        int lo = lower_bound_lds(skey, xs - EPSV);
        bool c = (hi - lo) >= MIN_SAMPLES;
        mask |= (unsigned)c << i;
        if (c) { cmax = fmaxf(cmax, xs); cminR = fminf(cminR, xs); }
    }

    // ---- block-wide exclusive prefix-max of core values (run_max) ----
    sbufF[tid] = cmax; __syncthreads();
    for (int o = 1; o < BS; o <<= 1) {
        float v = sbufF[tid];
        float w = (tid >= o) ? sbufF[tid - o] : -INF;
        __syncthreads();
        sbufF[tid] = fmaxf(v, w);
        __syncthreads();
    }
    float inMax = (tid > 0) ? sbufF[tid - 1] : -INF;
    __syncthreads();

    // ---- block-wide exclusive suffix-min of core values (right_val) ----
    sbufF[tid] = cminR; __syncthreads();
    for (int o = 1; o < BS; o <<= 1) {
        float v = sbufF[tid];
        float w = (tid + o < BS) ? sbufF[tid + o] : INF;
        __syncthreads();
        sbufF[tid] = fminf(v, w);
        __syncthreads();
    }
    float inMinR = (tid < BS - 1) ? sbufF[tid + 1] : INF;
    __syncthreads();

    // ---- count new clusters per chunk, then exclusive prefix-sum ----
    int cntLocal = 0;
    {
        float prev = inMax;
#pragma unroll
        for (int i = 0; i < CHUNK; ++i) {
            float xs = skey[g0i + i];
            bool c = (mask >> i) & 1;
            if (c && (xs - prev > EPSV)) ++cntLocal;
            if (c) prev = fmaxf(prev, xs);
        }
    }
    sbufI[tid] = cntLocal; __syncthreads();
    for (int o = 1; o < BS; o <<= 1) {
        int v = sbufI[tid];
        int w = (tid >= o) ? sbufI[tid - o] : 0;
        __syncthreads();
        sbufI[tid] = v + w;
        __syncthreads();
    }
    int total = sbufI[BS - 1];
    int inCnt = (tid > 0) ? sbufI[tid - 1] : 0;

    // ---- reverse walk: inclusive right_val within chunk (kept in VGPRs) ----
    float rv[CHUNK];
    {
        float run = inMinR;
#pragma unroll
        for (int i = CHUNK - 1; i >= 0; --i) {
            float xs = skey[g0i + i];
            bool c = (mask >> i) & 1;
            run = fminf(run, c ? xs : INF);
            rv[i] = run;
        }
    }

    // ---- forward walk: cluster ids + border attachment ----
    {
        float prev = inMax;                 // exclusive run_max (prev_core_val)
        int cnt = inCnt;
#pragma unroll
        for (int i = 0; i < CHUNK; ++i) {
            float xs = skey[g0i + i];
            bool c  = (mask >> i) & 1;
            bool nc = c && (xs - prev > EPSV);
            cnt += nc;
            int cid = cnt - 1;              // == left_cid (cid only changes at cores)
            float runIncl = c ? fmaxf(prev, xs) : prev;   // inclusive run_max = left_val
            float dl = xs - runIncl;
            float dr = rv[i] - xs;
            int rcid = cid + (((rv[i] - runIncl) > EPSV) ? 1 : 0);
            int bc   = (dl <= dr) ? cid : rcid;           // ties -> left
            bool bok = fminf(dl, dr) <= EPSV;
            slab[g0i + i] = c ? cid : (bok ? bc : -1);
            prev = runIncl;
        }
    }
    if (tid == 0) ncl[bd] = total;
    __syncthreads();

    // ---- unsort: each original point finds its sorted slot by binary search ----
    int* labRow = labels + (size_t)bd * NPTS;
    for (int n = tid; n < NPTS; n += BS) {
        float x = feat[((size_t)b * NPTS + n) * DIMS + d];
        int pos = lower_bound_lds(skey, x);
        labRow[n] = slab[pos];
    }
}

// Exclusive cumsum of cluster counts over the 64 dims of each batch.
__global__ void offsets_kernel(const int* __restrict__ ncl, int* __restrict__ off) {
    int t = threadIdx.x;                 // 512 = 8*64
    int b = t >> 6, d = t & 63;
    int s = 0;
    for (int k = 0; k < d; ++k) s += ncl[(b << 6) + k];
    off[t] = s;
}

__global__ void zero_kernel(f32x4* __restrict__ p, int n4) {
    int i = blockIdx.x * blockDim.x + threadIdx.x;
    int stride = gridDim.x * blockDim.x;
    f32x4 z = {0.f, 0.f, 0.f, 0.f};
    for (; i < n4; i += stride) p[i] = z;
}

// One thread per (b, d, n): disjoint one-hot stores (global ids per dim disjoint).
__global__ void scatter_kernel(const int* __restrict__ labels,
                               const int* __restrict__ off,
                               float* __restrict__ out) {
    int i = blockIdx.x * blockDim.x + threadIdx.x;   // < 512 * 16384
    int bd = i >> 14;
    int n  = i & (NPTS - 1);
    int lab = labels[i];
    if (lab < 0) return;
    int g = lab + off[bd];
    if (g >= NC_MAX) return;                         // overflow col sliced off in reference
    int b = bd >> 6;
    out[(((size_t)b << 14) + (size_t)n) * NC_MAX + (size_t)g] = 1.0f;
}

extern "C" void kernel_launch(void* const* d_in, const int* in_sizes, int n_in,
                              void* d_out, int out_size, void* d_ws, size_t ws_size,
                              hipStream_t stream) {
    (void)in_sizes; (void)n_in; (void)ws_size;
    const float* feat = (const float*)d_in[0];       // (8, 16384, 64) fp32
    float* out = (float*)d_out;                      // (8, 16384, 512) fp32

    int* labels = (int*)d_ws;                        // 512 * 16384 ints (32 MB)
    int* ncl    = labels + (size_t)BATCH * DIMS * NPTS;
    int* off    = ncl + BATCH * DIMS;

    const size_t shmem = (size_t)NPTS * 8 + 4096;    // keys + labels + scan buf = 132 KB

    dbscan1d_kernel<<<BATCH * DIMS, BS, shmem, stream>>>(feat, labels, ncl);
    offsets_kernel<<<1, BATCH * DIMS, 0, stream>>>(ncl, off);

    int n4 = out_size / 4;                           // 16,777,216 float4 stores
    zero_kernel<<<4096, 256, 0, stream>>>((f32x4*)out, n4);
    scatter_kernel<<<(BATCH * DIMS * NPTS) / 256, 256, 0, stream>>>(labels, off, out);
}